// ScannedRNN_3410204033434
// MI455X (gfx1250) — compile-verified
//
#include <hip/hip_runtime.h>

// GRU scan (T=1024, B=512, H=128) on gfx1250 (MI455X).
// - 32 independent workgroups (16 batch rows each), no cross-WG sync.
// - fp32 WMMA (v_wmma_f32_16x16x4_f32).
// - Wh B-fragments resident in VGPRs for all 1024 steps (64 VGPRs, no spills);
//   Wi transposed into LDS ([n][k], stride 132) -> B-fragments are single
//   immediate-offset ds_load_b64 (2addr-mergeable), conflict-free.
// - x tiles double-buffered in LDS via async global->LDS b128 copies
//   (ASYNCcnt), issued one step ahead, waited after the GEMM.
// - episode resets folded into accumulator-row scaling (no LDS RMW).

typedef __attribute__((ext_vector_type(2))) float v2f;
typedef __attribute__((ext_vector_type(4))) float v4f;
typedef __attribute__((ext_vector_type(8))) float v8f;

// Pointer types matching the async-to-LDS builtin's signature:
// param0: int4* in global (AS1) space, param1: int4* in LDS (AS3).
typedef int v4i __attribute__((vector_size(16)));
typedef __attribute__((address_space(1))) v4i* as1_v4i;
typedef __attribute__((address_space(3))) v4i* as3_v4i;

#define T_STEPS 1024
#define B_SIZE  512
#define H_SIZE  128
#define G3      384   // 3*H
#define NTHREADS 768  // 24 waves; one 16-col gate tile per wave
#define H_S 132       // padded LDS stride for h/x tiles (conflict-free A loads)
#define G_S 388       // padded LDS stride for gi/gh tiles
#define WT_S 132      // padded LDS stride for transposed Wi rows ([384][WT_S])

#if defined(__has_builtin)
#  if __has_builtin(__builtin_amdgcn_global_load_async_to_lds_b128)
#    define HAVE_ASYNC_B128 1
#  endif
#  if __has_builtin(__builtin_amdgcn_s_wait_asynccnt)
#    define HAVE_WAIT_ASYNC 1
#  endif
#endif

__device__ __forceinline__ void async_copy_b128(const float* g, float* l) {
#ifdef HAVE_ASYNC_B128
    // Integer round-trips: global generic == AS1 numerically; LDS generic
    // truncated to 32 bits == LDS offset (ISA aperture rule addr[31:0]).
    __builtin_amdgcn_global_load_async_to_lds_b128(
        (as1_v4i)(unsigned long long)g,
        (as3_v4i)(unsigned)(unsigned long long)l, 0, 0);
#else
    asm volatile("global_load_async_to_lds_b128 %0, %1, off"
                 :: "v"((unsigned)(unsigned long long)(const void*)l), "v"(g)
                 : "memory");
#endif
}

__device__ __forceinline__ void wait_async0() {
#ifdef HAVE_WAIT_ASYNC
    __builtin_amdgcn_s_wait_asynccnt(0);
#else
    asm volatile("s_wait_asynccnt 0x0" ::: "memory");
#endif
}

__device__ __forceinline__ float fast_tanh(float x) {
#if defined(__has_builtin) && __has_builtin(__builtin_amdgcn_tanhf)
    return __builtin_amdgcn_tanhf(x);   // CDNA5 V_TANH_F32 (trans op)
#else
    return tanhf(x);
#endif
}

__global__ __launch_bounds__(NTHREADS, 1)
void gru_scan_kernel(const float* __restrict__ ins,            // [T,B,H]
                     const unsigned char* __restrict__ resets, // [T,B] bool
                     const float* __restrict__ h0,             // [B,H]
                     const float* __restrict__ Wi,             // [H,3H]
                     const float* __restrict__ Wh,             // [H,3H]
                     const float* __restrict__ bi,             // [3H]
                     const float* __restrict__ bhn,            // [H]
                     float* __restrict__ out) {                // [T,B,H]
    extern __shared__ float smem[];
    float* s_wi  = smem;                    // [G3][WT_S] transposed Wi
    float* s_h   = s_wi + G3 * WT_S;        // [16][H_S]   hidden state tile
    float* s_x   = s_h  + 16 * H_S;         // [2][16][H_S] double-buffered x
    float* s_gi  = s_x  + 2 * 16 * H_S;     // [16][G_S]
    float* s_gh  = s_gi + 16 * G_S;         // [16][G_S]
    float* s_bi  = s_gh + 16 * G_S;         // [3H]
    float* s_bhn = s_bi + G3;               // [H]
    float* s_rm  = s_bhn + H_SIZE;          // [2][16] reset masks (0.0 / 1.0)

    const int tid  = threadIdx.x;
    const int lane = tid & 31;
    const int half = lane >> 4;             // K-pair selector within 16x4 frag
    const int l16  = lane & 15;             // M row (A) / N col (B,C,D)
    const int nidx = (tid >> 5) * 16 + l16; // this lane's gate column
    const int b0   = blockIdx.x * 16;       // batch tile base

    // ---- async-copy x tile for t=0 into buffer 0 (overlaps Wi transpose) ----
    if (tid < 512) {
        const int r = tid >> 5, cc = (tid & 31) * 4;   // 4 floats / b128
        async_copy_b128(ins + (size_t)(b0 + r) * H_SIZE + cc,
                        s_x + r * H_S + cc);
    }

    // ---- stage Wi transposed into LDS: s_wi[n][k] = Wi[k][n] ----
    // Coalesced global b128 reads, 4 scattered ds_store_b32 each (one-time).
    {
        const int k0 = tid / 96;            // 768 threads = 8 K-rows x 96 chunks
        const int c4 = (tid % 96) * 4;      // n base of this 4-wide chunk
#pragma unroll
        for (int i = 0; i < 16; ++i) {
            const int k = k0 + 8 * i;
            v4f w = *(const v4f*)(Wi + (size_t)k * G3 + c4);
            s_wi[(c4 + 0) * WT_S + k] = w[0];
            s_wi[(c4 + 1) * WT_S + k] = w[1];
            s_wi[(c4 + 2) * WT_S + k] = w[2];
            s_wi[(c4 + 3) * WT_S + k] = w[3];
        }
    }

    // ---- stage biases, h0 tile, reset mask for t=0 ----
    for (int i = tid; i < G3; i += NTHREADS) s_bi[i] = bi[i];
    if (tid < H_SIZE) s_bhn[tid] = bhn[tid];
    for (int i = tid; i < 16 * H_SIZE; i += NTHREADS) {
        int r = i >> 7, c = i & 127;
        s_h[r * H_S + c] = h0[(size_t)(b0 + r) * H_SIZE + c];
    }
    if (tid < 16) s_rm[tid] = resets[(size_t)b0 + tid] ? 0.0f : 1.0f;

    // ---- preload this wave's Wh B-fragments into VGPRs (held 1024 steps) ----
    // B 4x16 f32 fragment: lane = column nidx, K pair {4k+2*half, +1}.
    v2f Bwh[32];
#pragma unroll
    for (int k = 0; k < 32; ++k) {
        const int kk = 4 * k + 2 * half;
        v2f wh_f;
        wh_f.x = Wh[(size_t)kk * G3 + nidx];
        wh_f.y = Wh[(size_t)(kk + 1) * G3 + nidx];
        Bwh[k] = wh_f;
    }
    wait_async0();
    __syncthreads();

    // ---- incremental per-thread pointers (no per-step address rebuild) ----
    const int xr = tid >> 5, xc = (tid & 31) * 4;      // async-copy coords
    const float* xnext = ins + ((size_t)B_SIZE + b0 + xr) * H_SIZE + xc; // x_{t=1}
    const int yr = tid >> 5, yc0 = (tid & 31) * 4;     // gate-phase coords
    float* ydst = out + (size_t)(b0 + yr) * H_SIZE + yc0;
    const unsigned char* rnext = resets + (size_t)B_SIZE + b0;           // t=1 row

    const float* wrow = s_wi + nidx * WT_S; // lane-fixed base for Wi B-frags
    const v8f vzero = {};

    for (int t = 0; t < T_STEPS; ++t) {
        const int cur = t & 1, nxt = cur ^ 1;

        // ---- issue async copy of x_{t+1} (lands during this step's GEMM) ----
        if (t + 1 < T_STEPS && tid < 512)
            async_copy_b128(xnext, s_x + nxt * (16 * H_S) + xr * H_S + xc);

        // ---- GEMM phase: gi = x_t @ Wi(LDS), gh = h @ Wh(VGPR) ----
        const float* xrow = s_x + cur * (16 * H_S) + l16 * H_S;
        const float* hrow = s_h + l16 * H_S;
        v8f acc_i = vzero;
        v8f acc_h = vzero;
#pragma unroll
        for (int k = 0; k < 32; ++k) {
            const int kk = 4 * k + 2 * half;
            v2f a_x  = *(const v2f*)(xrow + kk);   // LDS b64, imm offset
            v2f a_h  = *(const v2f*)(hrow + kk);   // LDS b64, imm offset
            v2f b_wi = *(const v2f*)(wrow + kk);   // LDS b64, imm offset
            acc_i = __builtin_amdgcn_wmma_f32_16x16x4_f32(
                        false, a_x, false, b_wi,   (short)0, acc_i, false, false);
            acc_h = __builtin_amdgcn_wmma_f32_16x16x4_f32(
                        false, a_h, false, Bwh[k], (short)0, acc_h, false, false);
        }

        // ---- accumulators -> LDS; reset mask folded into gh rows ----
        // D layout: VGPR i = row (i + 8*half), col nidx.
#pragma unroll
        for (int i = 0; i < 8; ++i) {
            const int row = i + 8 * half;
            s_gi[row * G_S + nidx] = acc_i[i];
            s_gh[row * G_S + nidx] = acc_h[i] * s_rm[cur * 16 + row];
        }
        wait_async0();               // x_{t+1} landed (overlapped with GEMM)
        __syncthreads();

        // ---- gates + state update + streamed b128 output ----
        if (tid < 512) {
            const float* gi_r = s_gi + yr * G_S;
            const float* gh_r = s_gh + yr * G_S;
            const float  hm   = s_rm[cur * 16 + yr];
            v4f res;
#pragma unroll
            for (int j = 0; j < 4; ++j) {
                const int c = yc0 + j;
                float ir  = gi_r[c]       + s_bi[c];
                float iz  = gi_r[128 + c] + s_bi[128 + c];
                float inn = gi_r[256 + c] + s_bi[256 + c];
                float rg  = 1.0f / (1.0f + __expf(-(ir + gh_r[c])));
                float zg  = 1.0f / (1.0f + __expf(-(iz + gh_r[128 + c])));
                float ng  = fast_tanh(inn + rg * (gh_r[256 + c] + s_bhn[c]));
                float hp  = s_h[yr * H_S + c] * hm;    // reset-masked h_{t-1}
                res[j] = (1.0f - zg) * ng + zg * hp;
            }
            *(v4f*)(s_h + yr * H_S + yc0) = res;       // new hidden state
            *(v4f*)ydst = res;                          // b128 global store
        }
        // preload reset mask for t+1 into the alternate buffer
        if (tid < 16 && t + 1 < T_STEPS)
            s_rm[nxt * 16 + tid] = rnext[tid] ? 0.0f : 1.0f;
        __syncthreads();

        // advance incremental pointers
        xnext += (size_t)B_SIZE * H_SIZE;
        ydst  += (size_t)B_SIZE * H_SIZE;
        rnext += B_SIZE;
    }
}

extern "C" void kernel_launch(void* const* d_in, const int* in_sizes, int n_in,
                              void* d_out, int out_size, void* d_ws, size_t ws_size,
                              hipStream_t stream) {
    (void)in_sizes; (void)n_in; (void)out_size; (void)d_ws; (void)ws_size;
    const float*         ins    = (const float*)d_in[0];
    const unsigned char* resets = (const unsigned char*)d_in[1];
    const float*         h0     = (const float*)d_in[2];
    const float*         Wi     = (const float*)d_in[3];
    const float*         Wh     = (const float*)d_in[4];
    const float*         bi     = (const float*)d_in[5];
    const float*         bhn    = (const float*)d_in[6];
    float*               out    = (float*)d_out;

    const size_t lds_floats = (size_t)G3 * WT_S   // s_wi (transposed)
                            + 16 * H_S            // s_h
                            + 2 * 16 * H_S        // s_x (double buffer)
                            + 2 * 16 * G_S        // s_gi + s_gh
                            + G3 + H_SIZE + 32;   // biases + reset masks
    const size_t lds_bytes = lds_floats * sizeof(float); // ~273 KB (< 320 KB/WGP)

    gru_scan_kernel<<<dim3(B_SIZE / 16), dim3(NTHREADS), lds_bytes, stream>>>(
        ins, resets, h0, Wi, Wh, bi, bhn, out);
}